// MRFusionBlock_53687091200706
// MI455X (gfx1250) — compile-verified
//
#include <hip/hip_runtime.h>

typedef __attribute__((ext_vector_type(2))) float v2f;
typedef __attribute__((ext_vector_type(8))) float v8f;

#define CN0 300000
#define CN1 75000
#define CN2 18750

// ---------------------------------------------------------------------------
// GEMM: Z[M,N] = X[M,K] @ W[K,N] (+ bias), f32 via V_WMMA_F32_16X16X4_F32.
// One wave (32 threads) computes one 16x16 output tile.
// Both operands staged in LDS; inner loop is ds_load_2addr_b64 + v_wmma only.
//   A tile : [16][K+4] row-padded -> lane reads bank (4l + k0 + 2h) % 64, all
//            32 lanes distinct.
//   B strip: pair-interleaved [(k/2)*32 + 2*col + (k&1)] -> one b64 per lane,
//            banks {2l} U {2l+32}, conflict-free.
// Row guard hoisted into a wave-uniform fullTile branch: only the last row
// tile (M % 16 != 0) takes the guarded path.
// ---------------------------------------------------------------------------
template <int K>
__global__ __launch_bounds__(32) void gemm_wmma_f32(
    const float* __restrict__ X, const float* __restrict__ W,
    const float* __restrict__ bias, float* __restrict__ Z,
    int M, int N)
{
  __shared__ float Al[16 * (K + 4)];
  __shared__ float Blp[K * 16];

  const int tm   = blockIdx.x << 4;
  const int tn   = blockIdx.y << 4;
  const int lane = threadIdx.x;
  const int half = lane >> 4;               // 0: K pair {k0,k0+1}, 1: {k0+2,k0+3}
  const int l    = lane & 15;
  const bool fullTile = (tm + 16) <= M;     // wave-uniform

  // Speculative prefetch of the next row-tile of X (safe if OOB: dropped).
  __builtin_prefetch(X + ((size_t)tm + 16) * K + lane * 16, 0, 0);

  // Stage B strip W[K x 16], pair-interleaved for b64 operand reads.
  for (int i = lane; i < K * 16; i += 32) {
    const int k = i >> 4;
    const int c = i & 15;
    Blp[((k >> 1) << 5) + (c << 1) + (k & 1)] = W[k * N + tn + c];
  }

  // Stage A tile X[tm..tm+15, 0..K) with coalesced float4 loads.
  constexpr int NF4 = K >> 2;               // float4 per row
  if (fullTile) {
    for (int i = lane; i < 16 * NF4; i += 32) {
      const int r  = i / NF4;
      const int c4 = i % NF4;
      *(float4*)&Al[r * (K + 4) + (c4 << 2)] =
          ((const float4*)(X + (size_t)(tm + r) * K))[c4];
    }
  } else {
    for (int i = lane; i < 16 * NF4; i += 32) {
      const int r  = i / NF4;
      const int c4 = i % NF4;
      const int m  = tm + r;
      float4 v = {0.f, 0.f, 0.f, 0.f};
      if (m < M) v = ((const float4*)(X + (size_t)m * K))[c4];
      *(float4*)&Al[r * (K + 4) + (c4 << 2)] = v;
    }
  }
  __syncthreads();

  const float bv = bias ? bias[tn + l] : 0.0f;   // bias depends only on column
  v8f acc = {bv, bv, bv, bv, bv, bv, bv, bv};

  const int arow = l * (K + 4) + (half << 1);    // lane's A base (floats)

#pragma unroll 8
  for (int k0 = 0; k0 < K; k0 += 4) {
    const v2f a = *(const v2f*)&Al[arow + k0];
    const v2f b = *(const v2f*)&Blp[(((k0 >> 1) + half) << 5) | (l << 1)];
    acc = __builtin_amdgcn_wmma_f32_16x16x4_f32(false, a, false, b,
                                                (short)0, acc, false, false);
  }

  // D 16x16 f32: VGPR r -> row tm + r + 8*half, col tn + l
  if (fullTile) {
    float* __restrict__ zp = Z + (size_t)(tm + (half << 3)) * N + tn + l;
#pragma unroll
    for (int r = 0; r < 8; ++r) zp[(size_t)r * N] = acc[r];
  } else {
#pragma unroll
    for (int r = 0; r < 8; ++r) {
      const int m = tm + r + (half << 3);
      if (m < M) Z[(size_t)m * N + tn + l] = acc[r];
    }
  }
}

// ---------------------------------------------------------------------------
// Deterministic two-stage per-column mean/var -> (scale, shift)
// ---------------------------------------------------------------------------
__global__ void zero_f32(float* __restrict__ p, long long n) {
  long long i = (long long)blockIdx.x * blockDim.x + threadIdx.x;
  const long long stride = (long long)gridDim.x * blockDim.x;
  for (; i < n; i += stride) p[i] = 0.0f;
}

__global__ void col_stats_partial(const float* __restrict__ Z,
                                  float* __restrict__ part, int M, int N) {
  const int c = threadIdx.x;                 // blockDim.x == N
  float s = 0.f, s2 = 0.f;
  for (int m = blockIdx.x; m < M; m += gridDim.x) {
    const float z = Z[(size_t)m * N + c];
    s += z;
    s2 += z * z;
  }
  part[(size_t)blockIdx.x * (N << 1) + c] = s;
  part[(size_t)blockIdx.x * (N << 1) + N + c] = s2;
}

__global__ void col_stats_final(const float* __restrict__ part,
                                const float* __restrict__ g,
                                const float* __restrict__ beta,
                                float* __restrict__ ss, int NB, int M, int N) {
  const int c = threadIdx.x;                 // blockDim.x == N
  float s = 0.f, s2 = 0.f;
  for (int b = 0; b < NB; ++b) {             // fixed order -> deterministic
    s += part[(size_t)b * (N << 1) + c];
    s2 += part[(size_t)b * (N << 1) + N + c];
  }
  const float inv  = 1.0f / (float)M;
  const float mean = s * inv;
  const float var  = fmaxf(s2 * inv - mean * mean, 0.0f);
  const float scale = g[c] * rsqrtf(var + 1e-5f);
  ss[c]     = scale;
  ss[N + c] = beta[c] - mean * scale;
}

// out = relu(z*scale + shift), vectorized float4 (N is a multiple of 4)
__global__ void bn_relu(const float* __restrict__ Z, const float* __restrict__ ss,
                        float* __restrict__ out, long long tot4, int N) {
  const long long i = (long long)blockIdx.x * blockDim.x + threadIdx.x;
  if (i >= tot4) return;
  const int c = (int)((i << 2) & (long long)(N - 1));
  const float4 z = ((const float4*)Z)[i];
  float4 r;
  r.x = fmaxf(0.f, z.x * ss[c + 0] + ss[N + c + 0]);
  r.y = fmaxf(0.f, z.y * ss[c + 1] + ss[N + c + 1]);
  r.z = fmaxf(0.f, z.z * ss[c + 2] + ss[N + c + 2]);
  r.w = fmaxf(0.f, z.w * ss[c + 3] + ss[N + c + 3]);
  ((float4*)out)[i] = r;
}

// seg[cluster[m]] = max(seg, relu(...)); relu >= 0 so uint atomicMax on the
// bit pattern matches float max and is order-independent (deterministic).
__global__ void bn_relu_segmax(const float* __restrict__ Z,
                               const float* __restrict__ ss,
                               const int* __restrict__ cluster,
                               float* __restrict__ seg,
                               long long tot, int N, int shift) {
  const long long i = (long long)blockIdx.x * blockDim.x + threadIdx.x;
  if (i >= tot) return;
  const int c = (int)(i & (long long)(N - 1));
  const int m = (int)(i >> shift);
  const float v = fmaxf(0.f, Z[i] * ss[c] + ss[N + c]);
  atomicMax((unsigned int*)(seg + (size_t)cluster[m] * N + c),
            __float_as_uint(v));
}

// ---------------------------------------------------------------------------
// Fused residual outputs
// ---------------------------------------------------------------------------
__global__ void y0_fuse(const float* __restrict__ feat0,
                        const float* __restrict__ u10,
                        const float* __restrict__ u20,
                        const int* __restrict__ c01,
                        const int* __restrict__ c12,
                        float* __restrict__ y0) {
  const long long i = (long long)blockIdx.x * blockDim.x + threadIdx.x; // f4 idx
  if (i >= (long long)CN0 * 16) return;
  const int m  = (int)(i >> 4);
  const int c4 = (int)(i & 15);
  const int a = c01[m];
  const int b = c12[a];
  const float4 f = ((const float4*)feat0)[i];
  const float4 u = ((const float4*)(u10 + (size_t)a * 64))[c4];
  const float4 w = ((const float4*)(u20 + (size_t)b * 64))[c4];
  float4 r;
  r.x = fmaxf(0.f, f.x + u.x + w.x);
  r.y = fmaxf(0.f, f.y + u.y + w.y);
  r.z = fmaxf(0.f, f.z + u.z + w.z);
  r.w = fmaxf(0.f, f.w + u.w + w.w);
  ((float4*)y0)[i] = r;
}

__global__ void y1_fuse(const float* __restrict__ feat1,
                        const float* __restrict__ p1,
                        const float* __restrict__ u21,
                        const int* __restrict__ c12,
                        float* __restrict__ y1) {
  const long long i = (long long)blockIdx.x * blockDim.x + threadIdx.x; // f4 idx
  if (i >= (long long)CN1 * 32) return;
  const int m  = (int)(i >> 5);
  const int c4 = (int)(i & 31);
  const int a = c12[m];
  const float4 f = ((const float4*)feat1)[i];
  const float4 p = ((const float4*)p1)[i];
  const float4 u = ((const float4*)(u21 + (size_t)a * 128))[c4];
  float4 r;
  r.x = fmaxf(0.f, f.x + p.x + u.x);
  r.y = fmaxf(0.f, f.y + p.y + u.y);
  r.z = fmaxf(0.f, f.z + p.z + u.z);
  r.w = fmaxf(0.f, f.w + p.w + u.w);
  ((float4*)y1)[i] = r;
}

__global__ void y2_fuse(const float* __restrict__ feat2,
                        const float* __restrict__ s02,
                        const float* __restrict__ s12,
                        float* __restrict__ y2) {
  const long long i = (long long)blockIdx.x * blockDim.x + threadIdx.x; // f4 idx
  if (i >= (long long)CN2 * 64) return;
  const float4 f = ((const float4*)feat2)[i];
  const float4 a = ((const float4*)s02)[i];
  const float4 b = ((const float4*)s12)[i];
  float4 r;
  r.x = fmaxf(0.f, f.x + a.x + b.x);
  r.y = fmaxf(0.f, f.y + a.y + b.y);
  r.z = fmaxf(0.f, f.z + a.z + b.z);
  r.w = fmaxf(0.f, f.w + a.w + b.w);
  ((float4*)y2)[i] = r;
}

// ---------------------------------------------------------------------------
extern "C" void kernel_launch(void* const* d_in, const int* in_sizes, int n_in,
                              void* d_out, int out_size, void* d_ws, size_t ws_size,
                              hipStream_t stream) {
  const float* feat0 = (const float*)d_in[0];
  const float* feat1 = (const float*)d_in[1];
  const float* feat2 = (const float*)d_in[2];
  const int*   c01   = (const int*)d_in[3];
  const int*   c12   = (const int*)d_in[4];
  const float* up10_W = (const float*)d_in[5];
  const float* up10_b = (const float*)d_in[6];
  const float* up10_g = (const float*)d_in[7];
  const float* up10_be = (const float*)d_in[8];
  const float* up20_W = (const float*)d_in[9];
  const float* up20_b = (const float*)d_in[10];
  const float* up20_g = (const float*)d_in[11];
  const float* up20_be = (const float*)d_in[12];
  const float* up21_W = (const float*)d_in[13];
  const float* up21_b = (const float*)d_in[14];
  const float* up21_g = (const float*)d_in[15];
  const float* up21_be = (const float*)d_in[16];
  const float* dn01_W = (const float*)d_in[17];
  const float* dn01_g = (const float*)d_in[18];
  const float* dn01_be = (const float*)d_in[19];
  const float* dn02a_W = (const float*)d_in[20];
  const float* dn02a_g = (const float*)d_in[21];
  const float* dn02a_be = (const float*)d_in[22];
  const float* dn02b_W = (const float*)d_in[23];
  const float* dn02b_g = (const float*)d_in[24];
  const float* dn02b_be = (const float*)d_in[25];
  const float* dn12_W = (const float*)d_in[26];
  const float* dn12_g = (const float*)d_in[27];
  const float* dn12_be = (const float*)d_in[28];

  // Workspace layout (floats). Total ~75.9M floats (~304 MB).
  float* ws = (float*)d_ws;
  size_t off = 0;
  float* u10  = ws + off; off += (size_t)CN1 * 64;    // unpool stream1->0
  float* u20  = ws + off; off += (size_t)CN2 * 64;    // unpool stream2->0
  float* u21  = ws + off; off += (size_t)CN2 * 128;   // unpool stream2->1
  float* p1   = ws + off; off += (size_t)CN1 * 128;   // gridpool dn01
  float* tbuf = ws + off; off += (size_t)CN1 * 128;   // gridpool dn02a (= t)
  float* s02  = ws + off; off += (size_t)CN2 * 256;   // gridpool dn02b
  float* s12  = ws + off; off += (size_t)CN2 * 256;   // gridpool dn12
  float* zbig = ws + off; off += (size_t)CN0 * 128;   // reusable GEMM output
  float* part = ws + off; off += (size_t)256 * 512;   // stats partials
  float* ssb  = ws + off; off += 512;                 // per-column scale/shift

  float* y0 = (float*)d_out;
  float* y1 = y0 + (size_t)CN0 * 64;
  float* y2 = y1 + (size_t)CN1 * 128;

  const int NB = 256;

  auto run_gemm = [&](const float* X, const float* W, const float* b, float* Z,
                      int M, int K, int Nn) {
    dim3 grid((unsigned)((M + 15) / 16), (unsigned)(Nn / 16));
    if (K == 64)
      gemm_wmma_f32<64><<<grid, 32, 0, stream>>>(X, W, b, Z, M, Nn);
    else if (K == 128)
      gemm_wmma_f32<128><<<grid, 32, 0, stream>>>(X, W, b, Z, M, Nn);
    else
      gemm_wmma_f32<256><<<grid, 32, 0, stream>>>(X, W, b, Z, M, Nn);
  };
  auto run_stats = [&](const float* Z, const float* g, const float* be,
                       int M, int Nn) {
    col_stats_partial<<<NB, Nn, 0, stream>>>(Z, part, M, Nn);
    col_stats_final<<<1, Nn, 0, stream>>>(part, g, be, ssb, NB, M, Nn);
  };

  // Zero the four segment-max accumulators (contiguous region p1..s12).
  {
    long long nz = (long long)CN1 * 128 * 2 + (long long)CN2 * 256 * 2;
    zero_f32<<<1024, 256, 0, stream>>>(p1, nz);
  }

  // ---- unpool branches (linear+bias -> BN -> relu, stored densely) ----
  run_gemm(feat1, up10_W, up10_b, zbig, CN1, 128, 64);
  run_stats(zbig, up10_g, up10_be, CN1, 64);
  { long long t4 = (long long)CN1 * 64 / 4;
    bn_relu<<<(unsigned)((t4 + 255) / 256), 256, 0, stream>>>(zbig, ssb, u10, t4, 64); }

  run_gemm(feat2, up20_W, up20_b, zbig, CN2, 256, 64);
  run_stats(zbig, up20_g, up20_be, CN2, 64);
  { long long t4 = (long long)CN2 * 64 / 4;
    bn_relu<<<(unsigned)((t4 + 255) / 256), 256, 0, stream>>>(zbig, ssb, u20, t4, 64); }

  run_gemm(feat2, up21_W, up21_b, zbig, CN2, 256, 128);
  run_stats(zbig, up21_g, up21_be, CN2, 128);
  { long long t4 = (long long)CN2 * 128 / 4;
    bn_relu<<<(unsigned)((t4 + 255) / 256), 256, 0, stream>>>(zbig, ssb, u21, t4, 128); }

  // ---- gridpool branches (linear -> BN -> relu -> segment max) ----
  run_gemm(feat0, dn01_W, nullptr, zbig, CN0, 64, 128);
  run_stats(zbig, dn01_g, dn01_be, CN0, 128);
  { long long t = (long long)CN0 * 128;
    bn_relu_segmax<<<(unsigned)((t + 255) / 256), 256, 0, stream>>>(
        zbig, ssb, c01, p1, t, 128, 7); }

  run_gemm(feat0, dn02a_W, nullptr, zbig, CN0, 64, 128);
  run_stats(zbig, dn02a_g, dn02a_be, CN0, 128);
  { long long t = (long long)CN0 * 128;
    bn_relu_segmax<<<(unsigned)((t + 255) / 256), 256, 0, stream>>>(
        zbig, ssb, c01, tbuf, t, 128, 7); }

  run_gemm(tbuf, dn02b_W, nullptr, zbig, CN1, 128, 256);
  run_stats(zbig, dn02b_g, dn02b_be, CN1, 256);
  { long long t = (long long)CN1 * 256;
    bn_relu_segmax<<<(unsigned)((t + 255) / 256), 256, 0, stream>>>(
        zbig, ssb, c12, s02, t, 256, 8); }

  run_gemm(feat1, dn12_W, nullptr, zbig, CN1, 128, 256);
  run_stats(zbig, dn12_g, dn12_be, CN1, 256);
  { long long t = (long long)CN1 * 256;
    bn_relu_segmax<<<(unsigned)((t + 255) / 256), 256, 0, stream>>>(
        zbig, ssb, c12, s12, t, 256, 8); }

  // ---- fused residual outputs ----
  y0_fuse<<<(unsigned)(((long long)CN0 * 16 + 255) / 256), 256, 0, stream>>>(
      feat0, u10, u20, c01, c12, y0);
  y1_fuse<<<(unsigned)(((long long)CN1 * 32 + 255) / 256), 256, 0, stream>>>(
      feat1, p1, u21, c12, y1);
  y2_fuse<<<(unsigned)(((long long)CN2 * 64 + 255) / 256), 256, 0, stream>>>(
      feat2, s02, s12, y2);
}